// VqQuantizer_61933428408593
// MI455X (gfx1250) — compile-verified
//
#include <hip/hip_runtime.h>

// Problem constants (from reference)
#define T_TOKENS 65536          // 64*32*32
#define DIM      256
#define N_CODES  1024
#define ZQ_SIZE  (T_TOKENS * DIM)   // 16777216
#define LOSS_OFF ZQ_SIZE            // d_out[LOSS_OFF] = loss scalar
#define IDX_OFF  (ZQ_SIZE + 1)      // d_out[IDX_OFF + t] = (float)index[t]

typedef __attribute__((ext_vector_type(2))) float v2f;
typedef __attribute__((ext_vector_type(8))) float v8f;

__device__ __forceinline__ v8f wmma4(v2f a, v2f b, v8f c) {
    // D = A(16x4 f32) * B(4x16 f32) + C(16x16 f32)
    return __builtin_amdgcn_wmma_f32_16x16x4_f32(
        false, a, false, b, (short)0, c, false, false);
}

// ---- CDNA5 async LDS-load path (ASYNCcnt-tracked, overlaps with WMMA) ----
__device__ __forceinline__ void async_b128(unsigned lds_byte,
                                           const void* saddr,
                                           unsigned voff_bytes) {
    // GVS mode: mem = SADDR(64b SGPR pair) + VADDR(32b byte offset); VDST = LDS byte addr
    asm volatile("global_load_async_to_lds_b128 %0, %1, %2"
                 :: "v"(lds_byte), "v"(voff_bytes), "s"(saddr)
                 : "memory");
}

__device__ __forceinline__ void wait_async0() {
#if __has_builtin(__builtin_amdgcn_s_wait_asynccnt)
    __builtin_amdgcn_s_wait_asynccnt(0);
#else
    asm volatile("s_wait_asynccnt 0" ::: "memory");
#endif
}

__device__ __forceinline__ unsigned lds_off(const void* p) {
    // Generic LDS pointer: ADDR[31:0] is the LDS byte offset (aperture rule)
    return (unsigned)(unsigned long long)p;
}

// Prep: codebook row norms -> workspace; re-zero the loss slot each call
// (main kernel accumulates atomically; required for graph replay).
__global__ void __launch_bounds__(256) vq_prep(const float* __restrict__ cb,
                                               float* __restrict__ cbnorm,
                                               float* __restrict__ out) {
    int n = blockIdx.x * 256 + threadIdx.x;   // 0..1023
    const float4* row = (const float4*)(cb + (size_t)n * DIM);
    float s = 0.f;
#pragma unroll 8
    for (int i = 0; i < DIM / 4; ++i) {
        float4 v = row[i];
        s += v.x * v.x + v.y * v.y + v.z * v.z + v.w * v.w;
    }
    cbnorm[n] = s;
    if (n == 0) out[LOSS_OFF] = 0.f;
}

// Main: 512 blocks x 256 threads (8 wave32). Block owns 128 tokens.
__global__ void __launch_bounds__(256) vq_main(const float* __restrict__ z,
                                               const float* __restrict__ cb,
                                               const float* __restrict__ cbnorm,
                                               float* __restrict__ out) {
    __shared__ float zTile[128 * DIM];      // 128 KB
    __shared__ float cbTile[2][64 * DIM];   // 2 x 64 KB double buffer
    __shared__ float cbnormS[N_CODES];      //   4 KB
    __shared__ float znormS[128];
    __shared__ int   idxS[128];
    __shared__ float lossAcc;

    const int tid  = threadIdx.x;
    const int w    = tid >> 5;       // wave 0..7
    const int lane = tid & 31;
    const int half = lane >> 4;      // 0: K lo pair, 1: K hi pair
    const int r    = lane & 15;      // row within 16
    const int tokBase = blockIdx.x * 128;

    // ---- Prologue: async-stage z tile (128KB) + codebook chunk 0 (64KB) ----
    {
        const void* zsrc = (const void*)(z + (size_t)tokBase * DIM);
        unsigned zl = lds_off(zTile) + tid * 16;
        unsigned zv = tid * 16;
#pragma unroll
        for (int i = 0; i < 32; ++i) {          // 128KB / 256 threads / 16B
            async_b128(zl, zsrc, zv);
            zl += 4096; zv += 4096;
        }
        unsigned cl = lds_off(cbTile[0]) + tid * 16;
        unsigned cv = tid * 16;
#pragma unroll
        for (int i = 0; i < 16; ++i) {          // 64KB / 256 threads / 16B
            async_b128(cl, (const void*)cb, cv);
            cl += 4096; cv += 4096;
        }
        for (int i = tid; i < N_CODES; i += 256) cbnormS[i] = cbnorm[i];
        if (tid == 0) lossAcc = 0.f;
    }
    wait_async0();
    __syncthreads();

    if (tid < 128) {                   // ||z_t||^2 per token
        const float4* zr = (const float4*)(zTile + tid * DIM);
        float s = 0.f;
#pragma unroll 8
        for (int i = 0; i < DIM / 4; ++i) {
            float4 v = zr[i];
            s += v.x * v.x + v.y * v.y + v.z * v.z + v.w * v.w;
        }
        znormS[tid] = s;
    }

    // Running per-lane argmin state: token fixed per (v, lane-half)
    float best[8];
    int   bestn[8];
#pragma unroll
    for (int v = 0; v < 8; ++v) { best[v] = 3.4e38f; bestn[v] = 0x7fffffff; }

    // WMMA f32 16x16x4 per-lane layout: lane<16 -> (k0,k0+1), lane>=16 -> (k0+2,k0+3)
    const float* aBase = zTile + (w * 16 + r) * DIM + half * 2;
    const unsigned bLane = r * DIM + half * 2;

    // ---- Code chunk loop: 16 chunks x 64 codes, async double-buffered ----
    for (int nc = 0; nc < 16; ++nc) {
        const int n0 = nc * 64;
        if (nc > 0) {
            wait_async0();      // chunk nc resident in cbTile[nc&1]
            __syncthreads();    // all waves' portions done; prev buffer free
        }
        if (nc < 15) {          // prefetch chunk nc+1 into the idle buffer
            unsigned cl = lds_off(cbTile[(nc + 1) & 1]) + tid * 16;
            unsigned cv = (unsigned)((n0 + 64) * DIM * 4) + tid * 16;
#pragma unroll
            for (int i = 0; i < 16; ++i) {
                async_b128(cl, (const void*)cb, cv);
                cl += 4096; cv += 4096;
            }
        }

        const float* bBase = cbTile[nc & 1] + bLane;
        v8f c0 = {}, c1 = {}, c2 = {}, c3 = {};
#pragma unroll 8
        for (int k0 = 0; k0 < DIM; k0 += 4) {
            v2f a  = *(const v2f*)(aBase + k0);
            v2f b0 = *(const v2f*)(bBase + k0);
            v2f b1 = *(const v2f*)(bBase + 16 * DIM + k0);
            v2f b2 = *(const v2f*)(bBase + 32 * DIM + k0);
            v2f b3 = *(const v2f*)(bBase + 48 * DIM + k0);
            c0 = wmma4(a, b0, c0);
            c1 = wmma4(a, b1, c1);
            c2 = wmma4(a, b2, c2);
            c3 = wmma4(a, b3, c3);
        }

        // score = ||e||^2 - 2 z.e  (||z||^2 deferred to loss); ascending-n
        // scan + strict '<' preserves argmin first-occurrence semantics.
        const float cn0 = cbnormS[n0 +      r];
        const float cn1 = cbnormS[n0 + 16 + r];
        const float cn2 = cbnormS[n0 + 32 + r];
        const float cn3 = cbnormS[n0 + 48 + r];
#pragma unroll
        for (int v = 0; v < 8; ++v) {
            float s0 = fmaf(-2.f, c0[v], cn0);
            if (s0 < best[v]) { best[v] = s0; bestn[v] = n0 + r; }
            float s1 = fmaf(-2.f, c1[v], cn1);
            if (s1 < best[v]) { best[v] = s1; bestn[v] = n0 + 16 + r; }
            float s2 = fmaf(-2.f, c2[v], cn2);
            if (s2 < best[v]) { best[v] = s2; bestn[v] = n0 + 32 + r; }
            float s3 = fmaf(-2.f, c3[v], cn3);
            if (s3 < best[v]) { best[v] = s3; bestn[v] = n0 + 48 + r; }
        }
    }

    // ---- Final: reduce argmin across the 16 lanes of each half ----
    float wloss = 0.f;
#pragma unroll
    for (int v = 0; v < 8; ++v) {
        float d = best[v];
        int   n = bestn[v];
#pragma unroll
        for (int s = 1; s < 16; s <<= 1) {   // stays within each 16-lane half
            float od = __shfl_xor(d, s, 32);
            int   on = __shfl_xor(n, s, 32);
            if (od < d || (od == d && on < n)) { d = od; n = on; }
        }
        int tLoc = w * 16 + v + half * 8;    // token within block
        if (r == 0) {                        // lanes 0 and 16
            idxS[tLoc] = n;
            out[IDX_OFF + tokBase + tLoc] = (float)n;
            wloss += znormS[tLoc] + d;       // ||z-e*||^2
        }
    }
    if (r == 0) atomicAdd(&lossAcc, wloss);
    __syncthreads();

    // ---- z_q gather: wave copies its 16 tokens (2 float4 per lane/token) ----
    for (int t = 0; t < 16; ++t) {
        int tLoc = w * 16 + t;
        int n = idxS[tLoc];
        const float4* src = (const float4*)(cb + (size_t)n * DIM);
        float4* dst = (float4*)(out + (size_t)(tokBase + tLoc) * DIM);
        dst[lane]      = src[lane];
        dst[lane + 32] = src[lane + 32];
    }

    if (tid == 0)
        atomicAdd(out + LOSS_OFF, lossAcc * (11.0f / 16777216.0f)); // (1+beta)/count
}

extern "C" void kernel_launch(void* const* d_in, const int* in_sizes, int n_in,
                              void* d_out, int out_size, void* d_ws, size_t ws_size,
                              hipStream_t stream) {
    const float* z  = (const float*)d_in[0];   // [65536, 256] f32
    const float* cb = (const float*)d_in[1];   // [1024, 256] f32
    float* out      = (float*)d_out;           // zq | loss | indices(float)
    float* cbnorm   = (float*)d_ws;            // 1024 floats of scratch

    vq_prep<<<N_CODES / 256, 256, 0, stream>>>(cb, cbnorm, out);
    vq_main<<<T_TOKENS / 128, 256, 0, stream>>>(z, cb, cbnorm, out);
}